// GMP_55190329753896
// MI455X (gfx1250) — compile-verified
//
#include <hip/hip_runtime.h>
#include <math.h>

typedef float v2f __attribute__((ext_vector_type(2)));
typedef float v8f __attribute__((ext_vector_type(8)));

#define MEMB   8
#define DEG    5
#define WDIM   41            // 1 + DEG*MEMB
#define BATCH  32
#define LSEQ   4096
#define PCH    4128          // padded channel length (need >= 4111+8), mult of 32
#define NSP    4112          // padded s-range (4103 -> 257*16)
#define NTILE  257           // NSP/16
#define KF     48            // 40 real-part features + 8 imag features
#define CH_ELEMS (5*BATCH*PCH)
#define G_ELEMS  (BATCH*NSP*16)

// feature j -> LDS window slot (row-relative); folds to an immediate at compile time
__host__ __device__ constexpr int fcode(int j) {
  return (j < 40)
             ? ((((j % 5) == 1) ? 5 : (j % 5)) * 24 + (j / 5))  // d==1 -> const-1 row
             : (1 * 24 + (j - 40));                             // Im(xc) row
}

// ---------------- Stage 0: channel arrays -------------------------------
// CH[c][b][p]: c0 = Re(xc_pad), c1 = Im(xc_pad), c2 = amp, c3 = amp^2, c4 = amp^3
__global__ void k_channels(const float* __restrict__ x, float* __restrict__ CH) {
  int idx = blockIdx.x * blockDim.x + threadIdx.x;
  if (idx >= BATCH * PCH) return;
  int b = idx / PCH, p = idx - b * PCH;
  float xr = 0.f, xi = 0.f;
  int q = p - (2 * MEMB - 1);                 // 15 leading pad zeros
  if (q >= 0 && q < LSEQ) {
    xr = x[(b * LSEQ + q) * 2 + 0];
    xi = x[(b * LSEQ + q) * 2 + 1];
  }
  float amp = sqrtf(xr * xr + xi * xi);
  int o = b * PCH + p;
  CH[0 * BATCH * PCH + o] = xr;
  CH[1 * BATCH * PCH + o] = xi;
  CH[2 * BATCH * PCH + o] = amp;
  CH[3 * BATCH * PCH + o] = amp * amp;
  CH[4 * BATCH * PCH + o] = amp * amp * amp;
}

// ---------------- Stage 0b: build 48x16 B matrix from complex weights ----
// cols n<8 -> g_re[m=n], cols n>=8 -> g_im[m=n-8]
__global__ void k_bmat(const float* __restrict__ Wre, const float* __restrict__ Wim,
                       float* __restrict__ BM) {
  for (int e = threadIdx.x; e < KF * 16; e += 256) {
    int j = e >> 4, n = e & 15;
    float v;
    if (j < 40) {                      // real-part feature ch[d][s+m2], w = 1 + j
      int w = 1 + j;
      v = (n < 8) ? Wre[n * WDIM + w] : Wim[(n - 8) * WDIM + w];
    } else {                           // imag feature Im(xc)[s+m2], w = 1 + 5*m2
      int w = 1 + 5 * (j - 40);
      v = (n < 8) ? -Wim[n * WDIM + w] : Wre[(n - 8) * WDIM + w];
    }
    BM[e] = v;
  }
}

// ---------------- Stage 1: [rows x 48] @ [48 x 16] via V_WMMA_F32_16X16X4_F32
// one wave per 16-row tile; operands staged through LDS; all feature-index math
// folded to immediates (select on the lane half-bit only).
__global__ void __launch_bounds__(256) k_gemm(const float* __restrict__ CH,
                                              const float* __restrict__ BM,
                                              float* __restrict__ G) {
  __shared__ float sBM[KF * 16];       // 48x16 B matrix, shared by all 8 waves
  __shared__ float sWin[8][6 * 24];    // per-wave channel window; row 5 = const 1.0
  const int tid  = threadIdx.x;
  const int lane = tid & 31;
  const int wave = tid >> 5;
  const int gw   = blockIdx.x * 8 + wave;       // tile id, 0..8223
  const int b    = gw / NTILE;
  const int s0   = (gw - b * NTILE) * 16;
  const int half = lane >> 4;                   // K half: 0 = lanes 0-15, 1 = lanes 16-31
  const int mn   = lane & 15;                   // M for A, N for B/D
  const float* chb = CH + b * PCH;

  // stage B matrix (once per block) and the 23-wide channel window (per wave)
  for (int e = tid; e < KF * 16; e += 256) sBM[e] = BM[e];
  for (int i = lane; i < 6 * 24; i += 32) {
    int c = i / 24, pos = i - c * 24;           // pos 0..23; s0+23 <= 4119 < PCH
    sWin[wave][i] = (c < 5) ? chb[c * (BATCH * PCH) + (s0 + pos)] : 1.0f;
  }
  __syncthreads();

  const float* win = sWin[wave];
  v8f acc = {0.f, 0.f, 0.f, 0.f, 0.f, 0.f, 0.f, 0.f};
#pragma unroll
  for (int kt = 0; kt < 12; ++kt) {
    v2f a, bv;
#pragma unroll
    for (int v = 0; v < 2; ++v) {
      const int jlo = kt * 4 + v;               // lanes 0-15: K = jlo
      const int jhi = jlo + 2;                  // lanes 16-31: K = jhi
      int code = half ? fcode(jhi) : fcode(jlo);        // cndmask(imm, imm)
      int brow = jlo * 16 + half * 32;                  // (j*16) as const + half*32
      a[v]  = win[code + mn];
      bv[v] = sBM[brow + mn];
    }
    acc = __builtin_amdgcn_wmma_f32_16x16x4_f32(false, a, false, bv,
                                                (short)0, acc, false, false);
  }
  float* g = G + (b * NSP + s0) * 16;
#pragma unroll
  for (int r = 0; r < 8; ++r)                   // D layout: M = r + 8*half, N = mn
    g[(r + half * 8) * 16 + mn] = acc[r];
}

// ---------------- Stage 2: co[t] = sum_{b,m} xc[b,t+m]*(W[m,0]+g[b,m,t+m]),
// broadcast to all batch rows
__global__ void k_out(const float* __restrict__ CH, const float* __restrict__ G,
                      const float* __restrict__ Wre, const float* __restrict__ Wim,
                      float* __restrict__ out) {
  int t = blockIdx.x * blockDim.x + threadIdx.x;
  if (t >= LSEQ) return;
  float are = 0.f, aim = 0.f;
  for (int b = 0; b < BATCH; ++b) {
    const float* c0 = CH + b * PCH;
    const float* c1 = CH + BATCH * PCH + b * PCH;
    const float* gb = G + b * NSP * 16;
#pragma unroll
    for (int m = 0; m < MEMB; ++m) {
      int p = t + m;
      float xr = c0[p], xi = c1[p];
      float cr = Wre[m * WDIM] + gb[p * 16 + m];
      float ci = Wim[m * WDIM] + gb[p * 16 + 8 + m];
      are += xr * cr - xi * ci;
      aim += xr * ci + xi * cr;
    }
  }
  for (int b = 0; b < BATCH; ++b) {
    out[(b * LSEQ + t) * 2 + 0] = are;
    out[(b * LSEQ + t) * 2 + 1] = aim;
  }
}

extern "C" void kernel_launch(void* const* d_in, const int* in_sizes, int n_in,
                              void* d_out, int out_size, void* d_ws, size_t ws_size,
                              hipStream_t stream) {
  const float* x   = (const float*)d_in[0];
  // d_in[1] = h_0 (unused by the reference)
  const float* Wre = (const float*)d_in[2];
  const float* Wim = (const float*)d_in[3];
  float* ws = (float*)d_ws;
  float* CH = ws;                    // 5*32*4128 floats
  float* G  = CH + CH_ELEMS;         // 32*4112*16 floats
  float* BM = G + G_ELEMS;           // 48*16 floats   (~11.1 MB total)
  float* out = (float*)d_out;

  k_channels<<<(BATCH * PCH + 255) / 256, 256, 0, stream>>>(x, CH);
  k_bmat<<<1, 256, 0, stream>>>(Wre, Wim, BM);
  k_gemm<<<(BATCH * NTILE) / 8, 256, 0, stream>>>(CH, BM, G);   // 1028 blocks x 8 waves
  k_out<<<(LSEQ + 255) / 256, 256, 0, stream>>>(CH, G, Wre, Wim, out);
}